// LogLikelihoodLayer_51926154608729
// MI455X (gfx1250) — compile-verified
//
#include <hip/hip_runtime.h>
#include <hip/hip_bf16.h>
#include <math.h>

// ---------------------------------------------------------------------------
// Gaussian log-likelihood with full covariance, B=32, T=400, D=64.
// One wave32 workgroup per (b,t) matrix.
//  * sigma tile staged global->LDS via async direct-to-LDS copies (ASYNCcnt),
//    non-temporal (210MB stream, zero reuse, don't thrash 192MB L2)
//  * blocked (16-wide) right-looking Cholesky in LDS; trailing SYRK/GEMM
//    updates on the fp32 matrix pipe via v_wmma_f32_16x16x4_f32
//  * register-resident forward solve with wave32 shuffles
//  * deterministic fixed-order reduction (no float atomics)
// ---------------------------------------------------------------------------

typedef float v2f __attribute__((ext_vector_type(2)));
typedef float v8f __attribute__((ext_vector_type(8)));

#define LDS_STRIDE 68            // floats; 272B rows keep 16B alignment, mild bank conflicts
#define D_DIM      64
#define BT_TOTAL   (32 * 400)
#define EPS_F      1e-6f
// 0.5 * D * log(2*pi) for D=64
#define HALF_D_LOG_2PI 58.8120661250990500f

__global__ __launch_bounds__(32)
void chol_loglik_kernel(const float* __restrict__ x,
                        const float* __restrict__ mu,
                        const float* __restrict__ sigma,
                        float* __restrict__ partial)
{
    __shared__ float As[D_DIM * LDS_STRIDE];

    const int lane = threadIdx.x;       // 0..31
    const int bt   = blockIdx.x;        // 0..12799

    // ------- Stage sigma tile (64x64 = 16KB) straight into LDS (async) -----
    // Each lane copies 16B per issue; 32 issues cover the tile. Non-temporal:
    // every sigma byte is read exactly once by exactly one workgroup.
    {
        const float4* S4 = reinterpret_cast<const float4*>(
            sigma + (size_t)bt * (D_DIM * D_DIM));
        #pragma unroll
        for (int t = 0; t < 32; ++t) {
            int idx = t * 32 + lane;    // float4 index 0..1023
            int row = idx >> 4;         // 16 float4 per 64-float row
            int c4  = idx & 15;
            unsigned lds_addr =
                (unsigned)(uintptr_t)&As[row * LDS_STRIDE + c4 * 4];
            unsigned long long gaddr = (unsigned long long)(uintptr_t)(S4 + idx);
            asm volatile(
                "global_load_async_to_lds_b128 %0, %1, off th:TH_LOAD_NT"
                :: "v"(lds_addr), "v"(gaddr) : "memory");
        }
    }

    // Overlap: pull x - mu into registers while the async copies fly.
    const float* xp = x  + (size_t)bt * D_DIM;
    const float* mp = mu + (size_t)bt * D_DIM;
    float r0 = xp[lane]      - mp[lane];
    float r1 = xp[lane + 32] - mp[lane + 32];

    asm volatile("s_wait_asynccnt 0x0" ::: "memory");
    __syncthreads();

    // sigma + EPS*I
    As[lane        * (LDS_STRIDE + 1)] += EPS_F;
    As[(lane + 32) * (LDS_STRIDE + 1)] += EPS_F;
    __syncthreads();

    const int m = lane & 15;            // WMMA N / M-within-halfwave index
    const int h = lane >> 4;            // half-wave select

    // ---------------- Blocked Cholesky, panel width 16 ---------------------
    #pragma unroll
    for (int p = 0; p < 4; ++p) {
        const int d = p * 16;

        // (1) Factorize 16x16 diagonal block in-place (lanes 0..15 own rows).
        for (int j = 0; j < 16; ++j) {
            float ljj = sqrtf(As[(d + j) * LDS_STRIDE + d + j]);
            if (lane == j)
                As[(d + j) * LDS_STRIDE + d + j] = ljj;
            if (lane > j && lane < 16)
                As[(d + lane) * LDS_STRIDE + d + j] /= ljj;
            __syncthreads();
            if (lane > j && lane < 16) {
                float lij = As[(d + lane) * LDS_STRIDE + d + j];
                for (int k = j + 1; k <= lane; ++k)
                    As[(d + lane) * LDS_STRIDE + d + k] -=
                        lij * As[(d + k) * LDS_STRIDE + d + j];
            }
            __syncthreads();
        }

        if (p < 3) {
            // (2) Panel triangular solve: rows d+16..63, X * Lpp^T = A_panel.
            const int nr = 48 - d;
            for (int rr = lane; rr < nr; rr += 32) {
                const int R = d + 16 + rr;
                float xv[16];
                #pragma unroll
                for (int c = 0; c < 16; ++c) {
                    float acc = As[R * LDS_STRIDE + d + c];
                    #pragma unroll
                    for (int k = 0; k < c; ++k)
                        acc -= xv[k] * As[(d + c) * LDS_STRIDE + d + k];
                    xv[c] = acc / As[(d + c) * LDS_STRIDE + d + c];
                    As[R * LDS_STRIDE + d + c] = xv[c];
                }
            }
            __syncthreads();

            // (3) Trailing update A_ik -= L_ip * L_kp^T on the matrix pipe.
            //     f32 WMMA has no A/B NEG, so negate the A operand in VALU.
            #pragma unroll
            for (int i = p + 1; i < 4; ++i) {
                #pragma unroll
                for (int k = p + 1; k <= i; ++k) {
                    v8f c;
                    #pragma unroll
                    for (int v = 0; v < 8; ++v)
                        c[v] = As[(i * 16 + v + 8 * h) * LDS_STRIDE + k * 16 + m];
                    #pragma unroll
                    for (int k0 = 0; k0 < 16; k0 += 4) {
                        v2f a, b;
                        const int ar = (i * 16 + m) * LDS_STRIDE + d + k0 + 2 * h;
                        const int br = (k * 16 + m) * LDS_STRIDE + d + k0 + 2 * h;
                        a.x = -As[ar];
                        a.y = -As[ar + 1];
                        b.x =  As[br];
                        b.y =  As[br + 1];
                        c = __builtin_amdgcn_wmma_f32_16x16x4_f32(
                                false, a, false, b, (short)0, c, false, false);
                    }
                    #pragma unroll
                    for (int v = 0; v < 8; ++v)
                        As[(i * 16 + v + 8 * h) * LDS_STRIDE + k * 16 + m] = c[v];
                }
            }
            __syncthreads();
        }
    }

    // ---------------- log-det + forward solve + quadratic form -------------
    float ld = logf(As[lane        * (LDS_STRIDE + 1)]) +
               logf(As[(lane + 32) * (LDS_STRIDE + 1)]);
    #pragma unroll
    for (int off = 16; off > 0; off >>= 1)
        ld += __shfl_xor(ld, off, 32);

    float qf = 0.0f;
    for (int j = 0; j < 64; ++j) {
        float bj = (j < 32) ? __shfl(r0, j, 32) : __shfl(r1, j - 32, 32);
        float zj = bj / As[j * (LDS_STRIDE + 1)];
        qf += zj * zj;
        if (lane > j)       r0 -= As[lane        * LDS_STRIDE + j] * zj;
        if (lane + 32 > j)  r1 -= As[(lane + 32) * LDS_STRIDE + j] * zj;
    }

    if (lane == 0)
        partial[bt] = 0.5f * qf + ld + HALF_D_LOG_2PI;
}

// Deterministic fixed-order reduction (no float atomics).
__global__ __launch_bounds__(256)
void reduce_kernel(const float* __restrict__ partial, float* __restrict__ out, int n)
{
    __shared__ float s[256];
    float acc = 0.0f;
    for (int i = threadIdx.x; i < n; i += 256)
        acc += partial[i];
    s[threadIdx.x] = acc;
    __syncthreads();
    for (int off = 128; off > 0; off >>= 1) {
        if ((int)threadIdx.x < off)
            s[threadIdx.x] += s[threadIdx.x + off];
        __syncthreads();
    }
    if (threadIdx.x == 0)
        out[0] = s[0] / 32.0f;   // mean over batch (B=32)
}

extern "C" void kernel_launch(void* const* d_in, const int* in_sizes, int n_in,
                              void* d_out, int out_size, void* d_ws, size_t ws_size,
                              hipStream_t stream)
{
    const float* x     = (const float*)d_in[0];
    const float* mu    = (const float*)d_in[1];
    const float* sigma = (const float*)d_in[2];
    float* partial = (float*)d_ws;      // BT_TOTAL floats = 51200 bytes
    float* out     = (float*)d_out;

    chol_loglik_kernel<<<BT_TOTAL, 32, 0, stream>>>(x, mu, sigma, partial);
    reduce_kernel<<<1, 256, 0, stream>>>(partial, out, BT_TOTAL);
}